// DTMDecoder_1408749273326
// MI455X (gfx1250) — compile-verified
//
#include <hip/hip_runtime.h>
#include <math.h>

// Problem sizes (fixed by the reference)
#define BATCH  256
#define VOCAB  50000     // divisible by 16 -> no N-edge guards
#define NTOPIC 50
#define TSLICE 10
#define EMB    1024
#define TK     500       // TSLICE * NTOPIC
#define TKP    512       // padded to multiple of 16
#define NTILES (VOCAB / 16)   // 3125

typedef __attribute__((ext_vector_type(2))) float v2f;
typedef __attribute__((ext_vector_type(8))) float v8f;

// V_WMMA_F32_16X16X4_F32: D(16x16 f32) = A(16x4 f32) * B(4x16 f32) + C
__device__ __forceinline__ v8f wmma_f32_4(v2f a, v2f b, v8f c) {
  return __builtin_amdgcn_wmma_f32_16x16x4_f32(
      false, a, false, b, (short)0, c, false, false);
}

// ---------------------------------------------------------------------------
// K0: TEpad[512 x 1024] = TE with rows 500..511 zeroed (exact-zero padding so
// no masking is needed in the GEMM hot loop; WMMA propagates NaN, so padding
// must be real zeros).
// ---------------------------------------------------------------------------
__global__ __launch_bounds__(256) void tepad_kernel(
    const float* __restrict__ TE, float* __restrict__ TEpad) {
  const int i = blockIdx.x * 256 + threadIdx.x;   // < 512*1024
  TEpad[i] = (i < TK * EMB) ? TE[i] : 0.0f;
}

// ---------------------------------------------------------------------------
// K1: logits[m, n] = sum_e TEpad[m, e] * W[n, e]
// Block = 128 threads = 4 waves, all sharing one 16-col vocab tile (same B
// loads -> L0 hits). Each wave computes a 128x16 tile: 8 M-subtiles, so each
// K-step is 8 A loads (one base + immediate offsets) + 1 B load -> 8 WMMAs.
// A frag layout (32-bit 16x4): lane<16 holds {K,K+1} of row M=lane;
// lane>=16 holds {K+2,K+3}. B mirrors with lanes = N.
// ---------------------------------------------------------------------------
__global__ __launch_bounds__(128) void logits_gemm_kernel(
    const float* __restrict__ TEpad, const float* __restrict__ W,
    float* __restrict__ logits) {
  const int wave = threadIdx.x >> 5;   // 0..3 -> rows [wave*128, wave*128+127]
  const int lane = threadIdx.x & 31;
  const int half = lane >> 4;
  const int lm   = lane & 15;
  const int n0   = blockIdx.x * 16;    // 3125 blocks
  const int mBase = wave * 128;

  const float* Ap = TEpad + (size_t)(mBase + lm) * EMB + 2 * half;
  const float* Bp = W     + (size_t)(n0 + lm)    * EMB + 2 * half;

  v8f acc[8] = {};
  for (int k = 0; k < EMB; k += 4) {
    __builtin_prefetch(Bp + k + 256, 0, 3);        // global_prefetch_b8 on the
                                                   // HBM-streaming W operand
    v2f b = *(const v2f*)(Bp + k);
#pragma unroll
    for (int mt = 0; mt < 8; ++mt) {
      v2f a = *(const v2f*)(Ap + mt * (16 * EMB) + k);  // imm offset mt*64KB
      acc[mt] = wmma_f32_4(a, b, acc[mt]);
    }
  }

  // D layout: VGPR j -> row (tileRow0 + j + 8*half), col n0 + lm
#pragma unroll
  for (int mt = 0; mt < 8; ++mt) {
    float* outp =
        logits + (size_t)(mBase + mt * 16 + 8 * half) * VOCAB + n0 + lm;
#pragma unroll
    for (int j = 0; j < 8; ++j) outp[(size_t)j * VOCAB] = acc[mt][j];
  }
}

// ---------------------------------------------------------------------------
// K2: per-row max and sum(exp(x - max)) over VOCAB. One block per row.
// Padded rows are all-zero -> max 0, sum 50000 (unused downstream).
// ---------------------------------------------------------------------------
__global__ __launch_bounds__(256) void rowstats_kernel(
    const float* __restrict__ logits, float* __restrict__ rowmax,
    float* __restrict__ rowsum) {
  __shared__ float red[256];
  const int r   = blockIdx.x;            // 0..511
  const int tid = threadIdx.x;
  const float* row = logits + (size_t)r * VOCAB;

  float m = -3.4e38f;
  for (int i = tid; i < VOCAB; i += 256) m = fmaxf(m, row[i]);
  red[tid] = m; __syncthreads();
  for (int s = 128; s > 0; s >>= 1) {
    if (tid < s) red[tid] = fmaxf(red[tid], red[tid + s]);
    __syncthreads();
  }
  const float rm = red[0];
  __syncthreads();

  float sum = 0.0f;
  for (int i = tid; i < VOCAB; i += 256) sum += __expf(row[i] - rm);
  red[tid] = sum; __syncthreads();
  for (int s = 128; s > 0; s >>= 1) {
    if (tid < s) red[tid] += red[tid + s];
    __syncthreads();
  }
  if (tid == 0) { rowmax[r] = rm; rowsum[r] = red[0]; }
}

// ---------------------------------------------------------------------------
// K3: Wmat[b, c] with c = t*50 + k:
//     = theta[b,k] / rowsum[c] if t == time_index[b], else 0  (c >= 500 -> 0)
// ---------------------------------------------------------------------------
__global__ __launch_bounds__(256) void wmat_kernel(
    const float* __restrict__ theta, const int* __restrict__ tidx,
    const float* __restrict__ rowsum, float* __restrict__ wmat) {
  const int c = blockIdx.x * 256 + threadIdx.x;  // 0..511
  const int b = blockIdx.y;                      // 0..255
  float w = 0.0f;
  if (c < TK) {
    const int t = c / NTOPIC;
    const int k = c - t * NTOPIC;
    if (tidx[b] == t) w = theta[b * NTOPIC + k] / rowsum[c];
  }
  wmat[b * TKP + c] = w;
}

// ---------------------------------------------------------------------------
// K4: out(256 x 50000) = Wmat(256 x 512) * Beta(512 x 50000),
//     Beta[r, v] = exp(logits[r, v] - rowmax[r]) computed on the fly.
// One wave owns a full 256-row x 16-col strip (16 accumulator tiles), so each
// Beta element's exp is computed exactly once, and each K-step is
// 2 logits loads + 2 exp + 16 L2-resident wmat loads -> 16 WMMAs.
// Padded rows r>=500: logits==0, rowmax==0 -> Beta==1, weight==0 -> exact.
// ---------------------------------------------------------------------------
__global__ __launch_bounds__(128) void out_gemm_kernel(
    const float* __restrict__ logits, const float* __restrict__ rowmax,
    const float* __restrict__ wmat, float* __restrict__ out) {
  __shared__ float smax[TKP];
  for (int i = threadIdx.x; i < TKP; i += 128) smax[i] = rowmax[i];
  __syncthreads();

  const int wave  = threadIdx.x >> 5;
  const int lane  = threadIdx.x & 31;
  const int half  = lane >> 4;
  const int lm    = lane & 15;
  const int nTile = blockIdx.x * 4 + wave;
  if (nTile >= NTILES) return;          // wave-uniform early-out (after sync)
  const int n0 = nTile * 16;

  const float* Ap = wmat   + (size_t)lm * TKP + 2 * half;
  const float* Bp = logits + (size_t)(2 * half) * VOCAB + n0 + lm;

  v8f acc[16] = {};
  for (int kb = 0; kb < TKP; kb += 4) {
    const int r0 = kb + 2 * half;
    v2f b;
    b.x = __expf(Bp[(size_t)kb * VOCAB]         - smax[r0]);
    b.y = __expf(Bp[(size_t)kb * VOCAB + VOCAB] - smax[r0 + 1]);
#pragma unroll
    for (int mt = 0; mt < 16; ++mt) {
      v2f a = *(const v2f*)(Ap + mt * (16 * TKP) + kb);  // imm offset mt*32KB
      acc[mt] = wmma_f32_4(a, b, acc[mt]);
    }
  }

#pragma unroll
  for (int mt = 0; mt < 16; ++mt) {
    float* outp = out + (size_t)(mt * 16 + 8 * half) * VOCAB + n0 + lm;
#pragma unroll
    for (int j = 0; j < 8; ++j) outp[(size_t)j * VOCAB] = acc[mt][j];
  }
}

// ---------------------------------------------------------------------------
extern "C" void kernel_launch(void* const* d_in, const int* in_sizes, int n_in,
                              void* d_out, int out_size, void* d_ws,
                              size_t ws_size, hipStream_t stream) {
  const float* theta = (const float*)d_in[0];   // (256, 50)
  const float* W     = (const float*)d_in[1];   // (50000, 1024)
  const float* TE    = (const float*)d_in[2];   // (500, 1024)
  const int*   tidx  = (const int*)d_in[3];     // (256,)
  float* out = (float*)d_out;                   // (256, 50000)

  // Workspace (floats): logits[512*50000] | rowmax[512] | rowsum[512]
  //                     | wmat[256*512] | TEpad[512*1024]   (~105 MB)
  float* logits = (float*)d_ws;
  float* rowmax = logits + (size_t)TKP * VOCAB;
  float* rowsum = rowmax + TKP;
  float* wmat   = rowsum + TKP;
  float* TEpad  = wmat + (size_t)BATCH * TKP;

  // K0: build zero-padded TE copy (512*1024 elements, 256 threads/block)
  tepad_kernel<<<dim3((TKP * EMB) / 256), 256, 0, stream>>>(TE, TEpad);
  // K1: one block per 16-col vocab tile; 4 waves cover all 512 padded rows
  logits_gemm_kernel<<<dim3(NTILES), 128, 0, stream>>>(TEpad, W, logits);
  // K2: one block per padded row
  rowstats_kernel<<<dim3(TKP), 256, 0, stream>>>(logits, rowmax, rowsum);
  // K3: 256 batches x 512 columns
  wmat_kernel<<<dim3(2, BATCH), 256, 0, stream>>>(theta, tidx, rowsum, wmat);
  // K4: one wave per 16-col strip (full M), 4 waves/block, guard the tail
  out_gemm_kernel<<<dim3((NTILES + 3) / 4), 128, 0, stream>>>(logits, rowmax,
                                                              wmat, out);
}